// SPNET_82076825026567
// MI455X (gfx1250) — compile-verified
//
#include <hip/hip_runtime.h>
#include <hip/hip_bf16.h>

// ---------------------------------------------------------------------------
// GNN (edge-conv style) forward for MI455X / gfx1250.
// GEMMs use V_WMMA_F32_16X16X4_F32 (fp32 WMMA, wave32).
// Edge scatter uses global_atomic_add_f32; working set is L2-resident (192MB).
// ---------------------------------------------------------------------------

typedef __attribute__((ext_vector_type(2))) float v2f;
typedef __attribute__((ext_vector_type(8))) float v8f;

#define N_NODES 50000
#define N_EDGES 800000
#define IN_DIM  128
#define HIDDEN  64
#define OUT_DIM 10
#define N_LAYERS 3
#define N_GRAPHS 512
#define MT_PER_WAVE 5   // 50000/16 = 3125 = 5^5 m-tiles; 5 per wave -> 625 blocks

// ---------------------------------------------------------------------------
// Encoder GEMM: H = X @ W + bias   X:[N,128]  W:[128,64]  H:[N,64]
// 625 blocks x 4 waves; each wave: 5 m-tiles x 1 n-tile (16x16 each).
// B fragment loaded once per k-step, reused by 5 independent WMMA chains.
// ---------------------------------------------------------------------------
__global__ __launch_bounds__(128) void gemm_enc(
    const float* __restrict__ X, const float* __restrict__ W,
    const float* __restrict__ bias, float* __restrict__ H) {
  const int K = IN_DIM, Nc = HIDDEN;
  const int mbase = blockIdx.x * MT_PER_WAVE;  // first m-tile of this block
  const int wave  = threadIdx.x >> 5;          // n-tile 0..3
  const int lane  = threadIdx.x & 31;
  const int l16   = lane & 15;
  const int g     = lane >> 4;                 // half-wave group
  const int ncol  = wave * 16 + l16;           // B/C column for this lane

  const float* __restrict__ arow[MT_PER_WAVE];
#pragma unroll
  for (int t = 0; t < MT_PER_WAVE; ++t)
    arow[t] = X + (size_t)((mbase + t) * 16 + l16) * K;

  v8f acc[MT_PER_WAVE];
#pragma unroll
  for (int t = 0; t < MT_PER_WAVE; ++t) acc[t] = (v8f){};

#pragma unroll 4
  for (int k0 = 0; k0 < K; k0 += 4) {
    const int ka = k0 + 2 * g;
    v2f bf;
    bf.x = W[(size_t)ka * Nc + ncol];
    bf.y = W[(size_t)(ka + 1) * Nc + ncol];
#pragma unroll
    for (int t = 0; t < MT_PER_WAVE; ++t) {
      v2f af;
      af.x = arow[t][ka];
      af.y = arow[t][ka + 1];
      acc[t] = __builtin_amdgcn_wmma_f32_16x16x4_f32(false, af, false, bf,
                                                     (short)0, acc[t], false, false);
    }
  }
  const float bv = bias[ncol];
#pragma unroll
  for (int t = 0; t < MT_PER_WAVE; ++t) {
    float* __restrict__ out = H + (size_t)((mbase + t) * 16) * Nc + ncol;
#pragma unroll
    for (int r = 0; r < 8; ++r)
      out[(size_t)(r + 8 * g) * Nc] = acc[t][r] + bv;
  }
}

// ---------------------------------------------------------------------------
// Fused per-layer triple GEMM sharing A fragments (3 independent WMMA chains):
//   A   = H @ W1 + b1
//   B   = H @ W2
//   AGG = H @ W3 + b3    (init value for the edge scatter-add)
// ---------------------------------------------------------------------------
__global__ __launch_bounds__(128) void gemm3(
    const float* __restrict__ H,
    const float* __restrict__ W1, const float* __restrict__ b1,
    const float* __restrict__ W2,
    const float* __restrict__ W3, const float* __restrict__ b3,
    float* __restrict__ A, float* __restrict__ B, float* __restrict__ AGG) {
  const int K = HIDDEN, Nc = HIDDEN;
  const int mtile = blockIdx.x;
  const int wave  = threadIdx.x >> 5;
  const int lane  = threadIdx.x & 31;
  const int l16   = lane & 15;
  const int g     = lane >> 4;
  const int m     = mtile * 16 + l16;
  const int ncol  = wave * 16 + l16;

  const float* __restrict__ arow = H + (size_t)m * K;
  v8f c1 = {}, c2 = {}, c3 = {};
#pragma unroll 4
  for (int k0 = 0; k0 < K; k0 += 4) {
    const int ka = k0 + 2 * g;
    v2f af; af.x = arow[ka]; af.y = arow[ka + 1];
    v2f f1; f1.x = W1[(size_t)ka * Nc + ncol]; f1.y = W1[(size_t)(ka + 1) * Nc + ncol];
    v2f f2; f2.x = W2[(size_t)ka * Nc + ncol]; f2.y = W2[(size_t)(ka + 1) * Nc + ncol];
    v2f f3; f3.x = W3[(size_t)ka * Nc + ncol]; f3.y = W3[(size_t)(ka + 1) * Nc + ncol];
    c1 = __builtin_amdgcn_wmma_f32_16x16x4_f32(false, af, false, f1, (short)0, c1, false, false);
    c2 = __builtin_amdgcn_wmma_f32_16x16x4_f32(false, af, false, f2, (short)0, c2, false, false);
    c3 = __builtin_amdgcn_wmma_f32_16x16x4_f32(false, af, false, f3, (short)0, c3, false, false);
  }
  const float bv1 = b1[ncol];
  const float bv3 = b3[ncol];
  const size_t base = (size_t)(mtile * 16) * Nc + ncol;
#pragma unroll
  for (int r = 0; r < 8; ++r) {
    const size_t off = base + (size_t)(r + 8 * g) * Nc;
    A[off]   = c1[r] + bv1;
    B[off]   = c2[r];
    AGG[off] = c3[r] + bv3;
  }
}

// ---------------------------------------------------------------------------
// Edge message + scatter-add: agg[dst] += (a[src] - b[dst]) * ea
// 32 threads per edge, 2 features/thread (float2 gathers, L2-resident).
// ---------------------------------------------------------------------------
__global__ __launch_bounds__(256) void edge_msg(
    const int* __restrict__ src, const int* __restrict__ dst,
    const float* __restrict__ ea,
    const float* __restrict__ A, const float* __restrict__ B,
    float* __restrict__ AGG) {
  const int e  = blockIdx.x * 8 + (threadIdx.x >> 5);
  const int f2 = (threadIdx.x & 31) * 2;
  const int s = src[e];
  const int d = dst[e];
  const float w = ea[e];
  const float2 av = *(const float2*)(A + (size_t)s * HIDDEN + f2);
  const float2 bv = *(const float2*)(B + (size_t)d * HIDDEN + f2);
  float* aggp = AGG + (size_t)d * HIDDEN + f2;
  atomicAdd(aggp + 0, (av.x - bv.x) * w);
  atomicAdd(aggp + 1, (av.y - bv.y) * w);
}

__global__ __launch_bounds__(256) void relu_k(
    const float* __restrict__ AGG, float* __restrict__ H) {
  const size_t i = (size_t)blockIdx.x * 256 + threadIdx.x;
  H[i] = fmaxf(AGG[i], 0.0f);
}

__global__ __launch_bounds__(256) void zero_k(float* __restrict__ p, int n) {
  const int i = blockIdx.x * 256 + threadIdx.x;
  if (i < n) p[i] = 0.0f;
}

// Mean-pool accumulation: sums[batch[n]] += h[n], counts[batch[n]] += 1
__global__ __launch_bounds__(256) void pool_sum(
    const float* __restrict__ H, const int* __restrict__ batch,
    float* __restrict__ sums, float* __restrict__ cnts) {
  const int n = blockIdx.x * 4 + (threadIdx.x >> 6);
  const int f = threadIdx.x & 63;
  const int gidx = batch[n];
  atomicAdd(&sums[(size_t)gidx * HIDDEN + f], H[(size_t)n * HIDDEN + f]);
  if (f == 0) atomicAdd(&cnts[gidx], 1.0f);
}

// out[g,o] = (sums[g]/max(cnt,1)) @ fc_W + fc_b   (512x10, trivial)
__global__ __launch_bounds__(32) void pool_final(
    const float* __restrict__ sums, const float* __restrict__ cnts,
    const float* __restrict__ fcW, const float* __restrict__ fcb,
    float* __restrict__ out) {
  const int gidx = blockIdx.x;
  const int o = threadIdx.x;
  if (o >= OUT_DIM) return;
  float c = cnts[gidx];
  c = c > 1.0f ? c : 1.0f;
  const float inv = 1.0f / c;
  float acc = fcb[o];
#pragma unroll
  for (int k = 0; k < HIDDEN; ++k)
    acc += sums[(size_t)gidx * HIDDEN + k] * inv * fcW[k * OUT_DIM + o];
  out[gidx * OUT_DIM + o] = acc;
}

// ---------------------------------------------------------------------------
extern "C" void kernel_launch(void* const* d_in, const int* in_sizes, int n_in,
                              void* d_out, int out_size, void* d_ws, size_t ws_size,
                              hipStream_t stream) {
  const float* x     = (const float*)d_in[0];
  const int*   ei    = (const int*)d_in[1];      // [2, E] -> src row then dst row
  const float* ea    = (const float*)d_in[2];
  const int*   batch = (const int*)d_in[3];
  const float* W_enc = (const float*)d_in[4];
  const float* b_enc = (const float*)d_in[5];
  const float* W1    = (const float*)d_in[6];
  const float* b1    = (const float*)d_in[7];
  const float* W2    = (const float*)d_in[8];
  const float* W3    = (const float*)d_in[9];
  const float* b3    = (const float*)d_in[10];
  const float* fcW   = (const float*)d_in[11];
  const float* fcb   = (const float*)d_in[12];
  float* out = (float*)d_out;

  const int* src = ei;
  const int* dst = ei + N_EDGES;

  // Workspace layout (floats)
  float* ws   = (float*)d_ws;
  const size_t NH = (size_t)N_NODES * HIDDEN;           // 3.2M
  float* h    = ws;
  float* abuf = ws + NH;
  float* bbuf = ws + 2 * NH;
  float* agg  = ws + 3 * NH;
  float* sums = ws + 4 * NH;                            // 512*64
  float* cnts = sums + (size_t)N_GRAPHS * HIDDEN;       // 512

  const int MT = N_NODES / 16;                          // 3125 m-tiles

  // Encoder: h = x @ W_enc + b_enc  (5 m-tiles per wave)
  gemm_enc<<<MT / MT_PER_WAVE, 128, 0, stream>>>(x, W_enc, b_enc, h);

  for (int l = 0; l < N_LAYERS; ++l) {
    const float* W1l = W1 + (size_t)l * HIDDEN * HIDDEN;
    const float* W2l = W2 + (size_t)l * HIDDEN * HIDDEN;
    const float* W3l = W3 + (size_t)l * HIDDEN * HIDDEN;
    const float* b1l = b1 + (size_t)l * HIDDEN;
    const float* b3l = b3 + (size_t)l * HIDDEN;

    // a = h@W1+b1 ; b = h@W2 ; agg = h@W3+b3 (scatter-add init)
    gemm3<<<MT, 128, 0, stream>>>(h, W1l, b1l, W2l, W3l, b3l, abuf, bbuf, agg);
    // agg[dst] += (a[src] - b[dst]) * edge_attr
    edge_msg<<<N_EDGES / 8, 256, 0, stream>>>(src, dst, ea, abuf, bbuf, agg);
    // h = relu(agg)
    relu_k<<<(int)(NH / 256), 256, 0, stream>>>(agg, h);
  }

  // Global mean pool + final FC
  const int pool_elems = N_GRAPHS * HIDDEN + N_GRAPHS;  // sums + cnts contiguous
  zero_k<<<(pool_elems + 255) / 256, 256, 0, stream>>>(sums, pool_elems);
  pool_sum<<<N_NODES / 4, 256, 0, stream>>>(h, batch, sums, cnts);
  pool_final<<<N_GRAPHS, 32, 0, stream>>>(sums, cnts, fcW, fcb, out);
}